// SWD20_28449863369564
// MI455X (gfx1250) — compile-verified
//
#include <hip/hip_runtime.h>
#include <hip/hip_bf16.h>

// Problem constants (from reference): v is (B=8, L=4096, D=1024) f32,
// window W=128. q,k are unused.
#define L_DIM 4096
#define D_DIM 1024
#define W_DIM 128
#define TD    128   // d-channels per block == blockDim.x

#if defined(__has_builtin)
#  if __has_builtin(__builtin_amdgcn_global_load_async_to_lds_b32)
#    define HAS_ASYNC_LDS 1
#  endif
#endif
#ifndef HAS_ASYNC_LDS
#  define HAS_ASYNC_LDS 0
#endif

typedef __attribute__((address_space(1))) int* gptr_i32;
typedef __attribute__((address_space(3))) int* lptr_i32;

// ---- Bitonic network with all indices known at compile time --------------
// One (K, J) pass: 64 compare-exchanges with constant register indices.
template <int K, int J>
__device__ __forceinline__ void bitonic_pass(float (&x)[W_DIM]) {
#pragma unroll
  for (int i = 0; i < W_DIM; ++i) {
    const int ixj = i ^ J;              // J is constexpr -> constant after unroll
    if (ixj > i) {
      const bool up = ((i & K) == 0);   // K is constexpr -> folds
      const float a = x[i], c = x[ixj];
      const float lo = fminf(a, c);
      const float hi = fmaxf(a, c);
      x[i]   = up ? lo : hi;
      x[ixj] = up ? hi : lo;
    }
  }
}

template <int K, int J>
__device__ __forceinline__ void bitonic_j(float (&x)[W_DIM]) {
  bitonic_pass<K, J>(x);
  if constexpr (J > 1) bitonic_j<K, (J >> 1)>(x);
}

template <int K>
__device__ __forceinline__ void bitonic_k(float (&x)[W_DIM]) {
  bitonic_j<K, (K >> 1)>(x);
  if constexpr (K < W_DIM) bitonic_k<(K << 1)>(x);
}

__global__ __launch_bounds__(TD) void SWD20_sortwin_kernel(
    const float* __restrict__ v, float* __restrict__ out) {
  // smem layout: [l_local][d_local]; reads smem[i*TD + t] hit bank t%64
  // across a wave -> conflict-free; scatter writes likewise.
  __shared__ float smem[W_DIM * TD];   // 64 KB

  const int t    = threadIdx.x;            // d_local
  const int bx   = blockIdx.x;
  const int tile = bx & 7;                 // D/TD = 8
  const int w    = (bx >> 3) & 31;         // L/W  = 32
  const int b    = bx >> 8;                // B    = 8
  const int d0   = tile * TD;
  const int w0   = w * W_DIM;

  const float* __restrict__ vb = v + (size_t)b * L_DIM * D_DIM;

  // ---- Stage: diagonal gather-transpose into LDS -------------------------
  // element (row r, col d) belongs to window position l = (r + d) mod L.
  // For lane t (d = d0+t) and shift s, r = (w0 - d0 + s) mod L and the
  // window-local position is l_local = s + t, valid iff 0 <= s+t < 128.
  for (int s = -(W_DIM - 1); s <= (W_DIM - 1); ++s) {
    const int ll = s + t;
    if (0 <= ll && ll < W_DIM) {
      const int r = (w0 - d0 + s) & (L_DIM - 1);
      const float* gp = vb + (size_t)r * D_DIM + (d0 + t);   // coalesced per s
#if HAS_ASYNC_LDS
      // CDNA5 async copy: per-lane global addr -> per-lane LDS addr,
      // tracked with ASYNCcnt. Does the transpose-scatter for free.
      __builtin_amdgcn_global_load_async_to_lds_b32(
          (gptr_i32)gp,
          (lptr_i32)&smem[ll * TD + t],
          /*imm offset=*/0, /*cpol=*/0);
#else
      smem[ll * TD + t] = *gp;
#endif
    }
  }
#if HAS_ASYNC_LDS
  asm volatile("s_wait_asynccnt 0" ::: "memory");
#endif
  __syncthreads();

  // ---- Pull my column into registers ------------------------------------
  float x[W_DIM];
#pragma unroll
  for (int i = 0; i < W_DIM; ++i) x[i] = smem[i * TD + t];

  // ---- Fully unrolled bitonic sort (ascending), constant reg indices -----
  bitonic_k<2>(x);

  // ---- Coalesced store ---------------------------------------------------
  float* __restrict__ ob =
      out + (size_t)b * L_DIM * D_DIM + (size_t)w0 * D_DIM + (d0 + t);
#pragma unroll
  for (int i = 0; i < W_DIM; ++i) ob[(size_t)i * D_DIM] = x[i];
}

extern "C" void kernel_launch(void* const* d_in, const int* in_sizes, int n_in,
                              void* d_out, int out_size, void* d_ws, size_t ws_size,
                              hipStream_t stream) {
  (void)in_sizes; (void)n_in; (void)d_ws; (void)ws_size; (void)out_size;
  // inputs: d_in[0]=q, d_in[1]=k, d_in[2]=v  (only v is used by the reference)
  const float* v = (const float*)d_in[2];
  float* out = (float*)d_out;

  const int B = 8;
  const int blocks = B * (L_DIM / W_DIM) * (D_DIM / TD);   // 8*32*8 = 2048
  SWD20_sortwin_kernel<<<blocks, TD, 0, stream>>>(v, out);
}